// NeuralGaussianHSMMQCVV_44590350467624
// MI455X (gfx1250) — compile-verified
//
#include <hip/hip_runtime.h>
#include <math.h>

typedef __attribute__((ext_vector_type(2))) float v2f;
typedef __attribute__((ext_vector_type(8))) float v8f;

#define T_LEN 8192
#define K_ST  64
#define F_DIM 128
#define NCHUNK 256
#define CHLEN  32

// ---------------- prep: Gaussian coefficients ----------------
// Virtual coefficient matrix Grow[256][64]:
//   Grow[f][k]     = mu/var        (f < 128)
//   Grow[128+f][k] = -0.5/var
// Stored PAIR-INTERLEAVED for b64 B-fragment loads:
//   G[ ((f>>1)*64 + k)*2 + (f&1) ] = Grow[f][k]
// cvec[k] = -0.5 * sum_f ( log(2*pi*var) + mu^2/var )
__global__ void prep_coef(const float* __restrict__ means,
                          const float* __restrict__ log_vars,
                          float* __restrict__ G, float* __restrict__ cvec) {
  int k = blockIdx.x;
  int f = threadIdx.x;
  float var = fmaxf(expf(log_vars[k * F_DIM + f]), 1e-6f);
  float mu  = means[k * F_DIM + f];
  float inv = 1.0f / var;
  int fb = f;               // virtual row of b-coef
  int fa = F_DIM + f;       // virtual row of a-coef
  G[((fb >> 1) * K_ST + k) * 2 + (fb & 1)] = mu * inv;
  G[((fa >> 1) * K_ST + k) * 2 + (fa & 1)] = -0.5f * inv;
  float part = logf(6.2831853071795864f * var) + mu * mu * inv;
  __shared__ float red[F_DIM];
  red[f] = part;
  __syncthreads();
  for (int s = F_DIM / 2; s > 0; s >>= 1) {
    if (f < s) red[f] += red[f + s];
    __syncthreads();
  }
  if (f == 0) cvec[k] = -0.5f * red[0];
}

// lpi = log_softmax(log_pi)
__global__ void prep_pi(const float* __restrict__ log_pi, float* __restrict__ lpi) {
  int j = threadIdx.x;
  float m = -INFINITY;
  for (int i = 0; i < K_ST; ++i) m = fmaxf(m, log_pi[i]);
  float s = 0.0f;
  for (int i = 0; i < K_ST; ++i) s += expf(log_pi[i] - m);
  lpi[j] = log_pi[j] - m - logf(s);
}

// Aexp (row-stochastic + eps), stored pair-interleaved like G:
//   Ap[ ((i>>1)*64 + j)*2 + (i&1) ] = softmax(log_A[i,:])[j] + 1e-12
__global__ void prep_A(const float* __restrict__ log_A, float* __restrict__ Ap) {
  int i = threadIdx.x;
  float m = -INFINITY;
  for (int j = 0; j < K_ST; ++j) m = fmaxf(m, log_A[i * K_ST + j]);
  float s = 0.0f;
  for (int j = 0; j < K_ST; ++j) s += expf(log_A[i * K_ST + j] - m);
  float inv = 1.0f / s;
  for (int j = 0; j < K_ST; ++j)
    Ap[((i >> 1) * K_ST + j) * 2 + (i & 1)] = expf(log_A[i * K_ST + j] - m) * inv + 1e-12f;
}

// ---------------- emission GEMM (WMMA f32 16x16x4) ----------------
// emit[t,k] = cvec[k] + sum_{f<256} feat(t,f) * Grow[f][k]
// Writes E[t][k] = exp(emit - rowmax[t]), rowm[t] = rowmax.
__global__ void __launch_bounds__(128)
emission_wmma(const float* __restrict__ X, const float* __restrict__ G,
              const float* __restrict__ cvec, float* __restrict__ E,
              float* __restrict__ rowm) {
  int lane = threadIdx.x & 31;
  int w    = threadIdx.x >> 5;          // wave id: band of 16 rows
  int t0   = blockIdx.x * 64;           // block covers 64 timesteps
  int n    = lane & 15;
  int hi1  = lane >> 4;                 // 0 or 1
  int hi2  = hi1 << 1;                  // 0 or 2 : K-pair select
  int hi8  = hi1 << 3;                  // 0 or 8 : C/D row-half select
  int arow = t0 + 16 * w + n;           // A-operand row (time index)
  const float* xrow = X + arow * F_DIM;

  v8f acc[4] = {};                      // 16 x 64 output band
  // ---- linear features: fb = 4kk + hi2 in [0,128) ----
  #pragma unroll 8
  for (int kk = 0; kk < 32; ++kk) {
    int fb = 4 * kk + hi2;
    v2f a = *reinterpret_cast<const v2f*>(xrow + fb);
    int p = (fb >> 1) * K_ST;           // pair row base in G
    #pragma unroll
    for (int ct = 0; ct < 4; ++ct) {
      v2f b = *reinterpret_cast<const v2f*>(G + (p + 16 * ct + n) * 2);
      acc[ct] = __builtin_amdgcn_wmma_f32_16x16x4_f32(
          false, a, false, b, (short)0, acc[ct], false, false);
    }
  }
  // ---- squared features: fb = 128 + (4kk + hi2) ----
  #pragma unroll 8
  for (int kk = 0; kk < 32; ++kk) {
    int fl = 4 * kk + hi2;
    v2f x = *reinterpret_cast<const v2f*>(xrow + fl);
    v2f a; a.x = x.x * x.x; a.y = x.y * x.y;
    int p = ((F_DIM + fl) >> 1) * K_ST;
    #pragma unroll
    for (int ct = 0; ct < 4; ++ct) {
      v2f b = *reinterpret_cast<const v2f*>(G + (p + 16 * ct + n) * 2);
      acc[ct] = __builtin_amdgcn_wmma_f32_16x16x4_f32(
          false, a, false, b, (short)0, acc[ct], false, false);
    }
  }

  __shared__ float sEm[4][16][64];
  #pragma unroll
  for (int ct = 0; ct < 4; ++ct) {
    int col = 16 * ct + n;
    float cj = cvec[col];
    #pragma unroll
    for (int r = 0; r < 8; ++r)
      sEm[w][r + hi8][col] = acc[ct][r] + cj;
  }
  __syncthreads();

  int tid = threadIdx.x;
  if (tid < 64) {                       // one thread per timestep row
    int wv = tid >> 4, M = tid & 15;
    int t = t0 + 16 * wv + M;
    float m = -INFINITY;
    for (int j = 0; j < K_ST; ++j) m = fmaxf(m, sEm[wv][M][j]);
    rowm[t] = m;
    for (int j = 0; j < K_ST; ++j) E[t * K_ST + j] = expf(sEm[wv][M][j] - m);
  }
}

// ---------------- chunk-parallel scan (WMMA GEMM chain) ----------------
// Per chunk c: S := Identity; for t in chunk: S := (S * Aexp) col-scaled by E[t],
// row-band rescaled each step (log-scale accumulated per 16-row band).
__global__ void __launch_bounds__(128)
chunk_scan(const float* __restrict__ E, const float* __restrict__ rowm,
           const float* __restrict__ Ap, float* __restrict__ Mc,
           float* __restrict__ bs) {
  int c    = blockIdx.x;
  int lane = threadIdx.x & 31;
  int w    = threadIdx.x >> 5;          // band: rows [16w, 16w+16)
  int n    = lane & 15;
  int hi1  = lane >> 4;
  int hi2  = hi1 << 1;
  int hi8  = hi1 << 3;

  // B operands: Aexp resident in registers (16 ksteps x 4 col tiles), b64 loads
  v2f Bv[16][4];
  #pragma unroll
  for (int kk = 0; kk < 16; ++kk) {
    int p = (2 * kk + hi1) * K_ST;      // pair row (4kk+hi2)>>1
    #pragma unroll
    for (int ct = 0; ct < 4; ++ct)
      Bv[kk][ct] = *reinterpret_cast<const v2f*>(Ap + (p + 16 * ct + n) * 2);
  }

  // S band = identity rows
  v8f S[4];
  #pragma unroll
  for (int ct = 0; ct < 4; ++ct)
    #pragma unroll
    for (int r = 0; r < 8; ++r)
      S[ct][r] = ((16 * w + r + hi8) == (16 * ct + n)) ? 1.0f : 0.0f;

  __shared__ float sS[4][16][64];
  float bandLog = 0.0f, rms = 0.0f;
  int tstart = 1 + c * CHLEN;
  int tend = tstart + CHLEN; if (tend > T_LEN) tend = T_LEN;

  for (int t = tstart; t < tend; ++t) {
    // prefetch upcoming emission rows (global_prefetch_b8)
    int tp = (t + 4 < T_LEN) ? t + 4 : t;
    __builtin_prefetch(E + tp * K_ST, 0, 1);

    // stage band to LDS for D-layout -> A-operand transpose
    #pragma unroll
    for (int ct = 0; ct < 4; ++ct)
      #pragma unroll
      for (int r = 0; r < 8; ++r)
        sS[w][r + hi8][16 * ct + n] = S[ct][r];
    __syncthreads();

    v8f NS[4] = {};
    #pragma unroll
    for (int kk = 0; kk < 16; ++kk) {
      v2f a;
      a.x = sS[w][n][4 * kk + hi2];
      a.y = sS[w][n][4 * kk + hi2 + 1];
      #pragma unroll
      for (int ct = 0; ct < 4; ++ct)
        NS[ct] = __builtin_amdgcn_wmma_f32_16x16x4_f32(
            false, a, false, Bv[kk][ct], (short)0, NS[ct], false, false);
    }
    __syncthreads();

    // column-scale by emission likelihood; track band max
    float m = 0.0f;
    #pragma unroll
    for (int ct = 0; ct < 4; ++ct) {
      float e = E[t * K_ST + 16 * ct + n];
      #pragma unroll
      for (int r = 0; r < 8; ++r) {
        float v = NS[ct][r] * e;
        NS[ct][r] = v;
        m = fmaxf(m, v);
      }
    }
    #pragma unroll
    for (int off = 16; off > 0; off >>= 1)
      m = fmaxf(m, __shfl_xor(m, off, 32));
    m = fmaxf(m, 1e-30f);
    float inv = 1.0f / m;
    bandLog += logf(m);
    #pragma unroll
    for (int ct = 0; ct < 4; ++ct)
      #pragma unroll
      for (int r = 0; r < 8; ++r)
        S[ct][r] = NS[ct][r] * inv;
    rms += rowm[t];
  }

  #pragma unroll
  for (int ct = 0; ct < 4; ++ct)
    #pragma unroll
    for (int r = 0; r < 8; ++r)
      Mc[c * 4096 + (16 * w + r + hi8) * 64 + 16 * ct + n] = S[ct][r];
  if (lane == 0) bs[c * 4 + w] = bandLog + rms;
}

// ---------------- final combine: v0 through NCHUNK chunk operators ----------------
__global__ void __launch_bounds__(64)
combine(const float* __restrict__ E, const float* __restrict__ rowm,
        const float* __restrict__ lpi, const float* __restrict__ Mc,
        const float* __restrict__ bs, float* __restrict__ out) {
  __shared__ float sv[64];
  __shared__ float sw[64];
  int j = threadIdx.x;
  float lv = lpi[j] + logf(E[j] + 1e-37f);   // dp0 (minus rowm[0], added to logZ)
  sv[j] = lv;
  __syncthreads();
  float m = -INFINITY;
  for (int i = 0; i < 64; ++i) m = fmaxf(m, sv[i]);
  __syncthreads();
  float vj = expf(lv - m);
  sv[j] = vj;
  __syncthreads();
  float s = 0.0f;
  for (int i = 0; i < 64; ++i) s += sv[i];
  float logZ = m + logf(s) + rowm[0];
  float v = vj / s;
  __syncthreads();

  for (int c = 0; c < NCHUNK; ++c) {
    float b0 = bs[c * 4 + 0], b1 = bs[c * 4 + 1];
    float b2 = bs[c * 4 + 2], b3 = bs[c * 4 + 3];
    float mb = fmaxf(fmaxf(b0, b1), fmaxf(b2, b3));
    float sc = expf(((j >> 4) == 0 ? b0 : (j >> 4) == 1 ? b1 : (j >> 4) == 2 ? b2 : b3) - mb);
    sv[j] = v * sc;
    __syncthreads();
    float acc = 0.0f;
    const float* M = Mc + c * 4096;
    for (int a = 0; a < 64; ++a) acc += sv[a] * M[a * 64 + j];
    sw[j] = acc;
    __syncthreads();
    float ssum = 0.0f;
    for (int i = 0; i < 64; ++i) ssum += sw[i];
    ssum = fmaxf(ssum, 1e-37f);
    v = acc / ssum;
    logZ += mb + logf(ssum);
    __syncthreads();
  }
  if (j == 0) out[0] = logZ;
}

extern "C" void kernel_launch(void* const* d_in, const int* in_sizes, int n_in,
                              void* d_out, int out_size, void* d_ws, size_t ws_size,
                              hipStream_t stream) {
  const float* X        = (const float*)d_in[0];
  const float* log_pi   = (const float*)d_in[1];
  const float* log_A    = (const float*)d_in[2];
  const float* means    = (const float*)d_in[3];
  const float* log_vars = (const float*)d_in[4];

  float* ws   = (float*)d_ws;
  float* G    = ws;                 // 256*64        = 16384 (pair-interleaved)
  float* cvec = ws + 16384;         // 64
  float* lpi  = ws + 16448;         // 64
  float* Ap   = ws + 16512;         // 64*64         = 4096 (pair-interleaved)
  float* E    = ws + 20608;         // 8192*64       = 524288
  float* rowm = ws + 544896;        // 8192
  float* Mc   = ws + 553088;        // 256*64*64     = 1048576
  float* bs   = ws + 1601664;       // 256*4         = 1024
  // total: 1602688 floats ~= 6.4 MB of workspace

  prep_coef<<<64, 128, 0, stream>>>(means, log_vars, G, cvec);
  prep_pi<<<1, 64, 0, stream>>>(log_pi, lpi);
  prep_A<<<1, 64, 0, stream>>>(log_A, Ap);
  emission_wmma<<<128, 128, 0, stream>>>(X, G, cvec, E, rowm);
  chunk_scan<<<NCHUNK, 128, 0, stream>>>(E, rowm, Ap, Mc, bs);
  combine<<<1, 64, 0, stream>>>(E, rowm, lpi, Mc, bs, (float*)d_out);
}